// IntramodalSupCon_69621419868187
// MI455X (gfx1250) — compile-verified
//
#include <hip/hip_runtime.h>
#include <math.h>

typedef __attribute__((ext_vector_type(2))) float v2f;
typedef __attribute__((ext_vector_type(4))) float v4f;
typedef __attribute__((ext_vector_type(8))) float v8f;

#define NSAMP   4096      // N
#define DDIM    128       // D
#define MTOT    8192      // M = 2*N
#define PANEL   32        // columns per LDS panel (double buffered)
#define LSTRIDE 132       // padded float stride per column -> no 16-way bank conflict
#define INV_T   (1.0f / 0.07f)

// ---- async global->LDS staging support (gfx1250), with safe fallback --------
#if defined(__has_builtin)
#if __has_builtin(__builtin_amdgcn_global_load_async_to_lds_b128)
#define ASYNC_STAGE 1
#endif
#endif
#ifndef ASYNC_STAGE
#define ASYNC_STAGE 0
#endif

// Exact pointee type from the builtin's signature: int __vector_size__(16)
typedef int v4i __attribute__((__vector_size__(16)));
typedef __attribute__((address_space(1))) v4i *gv4i_ptr_t;   // global (__device__)
typedef __attribute__((address_space(3))) v4i *lv4i_ptr_t;   // LDS (__shared__)

__device__ __forceinline__ void stage_chunk(float* lds_dst, const float* gsrc) {
#if ASYNC_STAGE
    // AMDGPU flat->local addrspacecast == low-32-bit truncation of the flat addr
    __builtin_amdgcn_global_load_async_to_lds_b128(
        (gv4i_ptr_t)(uintptr_t)gsrc,
        (lv4i_ptr_t)(uintptr_t)(unsigned)(uintptr_t)lds_dst,
        0, 0);
#else
    *(v4f*)lds_dst = *(const v4f*)gsrc;
#endif
}

__device__ __forceinline__ void wait_stage() {
#if ASYNC_STAGE
#if __has_builtin(__builtin_amdgcn_s_wait_asynccnt)
    __builtin_amdgcn_s_wait_asynccnt(0);
#else
    asm volatile("s_wait_asynccnt 0x0" ::: "memory");
#endif
#endif
}

// flat[i][k] = embeds[i % N][i / N][k]  (view-major flatten)
__device__ __forceinline__ const float* flat_row(const float* embeds, int i) {
    return embeds + (size_t)(i & (NSAMP - 1)) * (2 * DDIM) + (size_t)(i >> 12) * DDIM;
}

// Stage one 32-column x 128-float panel; 256 threads x (PANEL/8) float4 chunks
__device__ __forceinline__ void stage_panel(float* buf, const float* embeds, int jp, int tid) {
    #pragma unroll
    for (int it = 0; it < PANEL / 8; ++it) {
        int idx = tid + it * 256;          // 0 .. PANEL*32-1
        int col = idx >> 5;                // 32 float4s per column
        int k4  = idx & 31;
        stage_chunk(&buf[col * LSTRIDE + k4 * 4], flat_row(embeds, jp + col) + k4 * 4);
    }
}

__global__ __launch_bounds__(256) void supcon_rows_kernel(
    const float* __restrict__ embeds,
    const long long* __restrict__ labels,
    float* __restrict__ rowloss)
{
    __shared__ float panel[2][PANEL * LSTRIDE];

    const int tid  = threadIdx.x;
    const int wave = tid >> 5;       // 0..7
    const int lane = tid & 31;
    const int half = lane >> 4;      // 0 or 1
    const int l16  = lane & 15;

    // This wave owns 16 rows [rowBase, rowBase+16)
    const int rowBase = blockIdx.x * 128 + wave * 16;

    // ---- Preload A fragments: lane holds row rowBase+l16, k = 4*kk + 2*half + {0,1}
    const float* aptr = flat_row(embeds, rowBase + l16);
    v2f afrag[32];
    #pragma unroll
    for (int kk = 0; kk < 32; ++kk) {
        const float* p = aptr + kk * 4 + half * 2;
        v2f a; a.x = p[0]; a.y = p[1];
        afrag[kk] = a;
    }

    // ---- Row labels for the 8 rows this lane sees in the C layout
    int rowlab[8];
    #pragma unroll
    for (int r = 0; r < 8; ++r) {
        int irow = rowBase + r + 8 * half;
        rowlab[r] = (int)labels[irow & (NSAMP - 1)];
    }

    // ---- Per-lane online-softmax state for 8 rows
    float rm[8], rd[8], sm[8], cn[8];
    #pragma unroll
    for (int r = 0; r < 8; ++r) { rm[r] = -INFINITY; rd[r] = 0.f; sm[r] = 0.f; cn[r] = 0.f; }

    // ---- Prologue: stage panel 0
    stage_panel(panel[0], embeds, 0, tid);
    wait_stage();
    __syncthreads();

    const int nP = MTOT / PANEL;
    for (int p = 0; p < nP; ++p) {
        const int jp = p * PANEL;
        const float* cur = panel[p & 1];

        // Issue async staging of the next panel into the other buffer
        if (p + 1 < nP)
            stage_panel(panel[(p + 1) & 1], embeds, jp + PANEL, tid);
        // L2 prefetch of the panel after next (cacheline granular)
        if (p + 2 < nP && tid < PANEL * 4)
            __builtin_prefetch(flat_row(embeds, jp + 2 * PANEL + (tid >> 2)) + (tid & 3) * 32, 0, 1);

        #pragma unroll
        for (int st = 0; st < PANEL / 16; ++st) {
            const int jt = jp + st * 16;
            v8f c = {};
            const float* bbase = &cur[(st * 16 + l16) * LSTRIDE + half * 2];
            #pragma unroll
            for (int kk = 0; kk < 32; ++kk) {
                v2f b = *(const v2f*)(bbase + kk * 4);
                c = __builtin_amdgcn_wmma_f32_16x16x4_f32(
                        /*neg_a=*/false, afrag[kk],
                        /*neg_b=*/false, b,
                        /*c_mod=*/(short)0, c,
                        /*reuse_a=*/false, /*reuse_b=*/false);
            }

            // Epilogue: online max / denom / masked-sum update per element
            const int jcol   = jt + l16;
            const int collab = (int)labels[jcol & (NSAMP - 1)];
            #pragma unroll
            for (int r = 0; r < 8; ++r) {
                float x    = c[r] * INV_T;
                int   irow = rowBase + r + 8 * half;
                bool  selfp = (irow == jcol);
                float newm = fmaxf(rm[r], x);
                float eold = __expf(rm[r] - newm);
                float enew = selfp ? 0.f : __expf(x - newm);
                rd[r] = rd[r] * eold + enew;
                rm[r] = newm;
                bool pos = (rowlab[r] == collab) && !selfp;
                sm[r] += pos ? x : 0.f;
                cn[r] += pos ? 1.f : 0.f;
            }
        }

        // Next panel must be fully landed in LDS before anyone reads it;
        // also fences reads of `cur` before the next iteration's staging overwrites it.
        wait_stage();
        __syncthreads();
    }

    // ---- Reduce each row's state across its 16 lanes (flash merge for (m,d))
    #pragma unroll
    for (int r = 0; r < 8; ++r) {
        float m = rm[r], d = rd[r], s = sm[r], c = cn[r];
        #pragma unroll
        for (int off = 8; off > 0; off >>= 1) {   // xor 8,4,2,1 stays within 16-lane half
            float m2 = __shfl_xor(m, off, 32);
            float d2 = __shfl_xor(d, off, 32);
            float s2 = __shfl_xor(s, off, 32);
            float c2 = __shfl_xor(c, off, 32);
            float nm = fmaxf(m, m2);
            d = d * __expf(m - nm) + d2 * __expf(m2 - nm);
            m = nm; s += s2; c += c2;
        }
        if (l16 == 0) {
            int irow = rowBase + r + 8 * half;
            // mean_log_prob_i = sum_mask(logits)/cnt - m_i - log(denom_i)
            float mlp = s / c - m - __logf(d);
            rowloss[irow] = -mlp;
        }
    }
}

__global__ __launch_bounds__(256) void supcon_reduce_kernel(
    const float* __restrict__ rowloss, float* __restrict__ out)
{
    __shared__ float sbuf[256];
    float acc = 0.f;
    for (int i = threadIdx.x; i < MTOT; i += 256)
        acc += rowloss[i];
    sbuf[threadIdx.x] = acc;
    __syncthreads();
    for (int off = 128; off > 0; off >>= 1) {
        if (threadIdx.x < off) sbuf[threadIdx.x] += sbuf[threadIdx.x + off];
        __syncthreads();
    }
    if (threadIdx.x == 0) out[0] = sbuf[0] / (float)MTOT;
}

extern "C" void kernel_launch(void* const* d_in, const int* in_sizes, int n_in,
                              void* d_out, int out_size, void* d_ws, size_t ws_size,
                              hipStream_t stream) {
    const float*     embeds = (const float*)d_in[0];       // (4096, 2, 128) f32
    const long long* labels = (const long long*)d_in[1];   // (4096,) i64
    float* rowloss = (float*)d_ws;                         // 8192 floats of scratch
    float* out     = (float*)d_out;                        // scalar f32 loss

    dim3 grid(MTOT / 128);   // 64 blocks, 8 waves each, 16 rows per wave
    dim3 block(256);
    supcon_rows_kernel<<<grid, block, 0, stream>>>(embeds, labels, rowloss);
    supcon_reduce_kernel<<<1, 256, 0, stream>>>(rowloss, out);
}